// LanczosStepTorch_69896297775455
// MI455X (gfx1250) — compile-verified
//
#include <hip/hip_runtime.h>
#include <math.h>

// ---------------------------------------------------------------------------
// Lanczos step for MI455X (gfx1250, wave32).
//   ws layout (floats): [0]=d  [1]=d2  [2..21]=w[0..19]  [22]=ss  [23]=scale
//                       [24..43]=h[0..19]
//   d_out layout: b[0..N) | T[N..N+1089) | beta[N+1089]
// ---------------------------------------------------------------------------

typedef __attribute__((ext_vector_type(2))) float v2f;
typedef __attribute__((ext_vector_type(8))) float v8f;

__device__ __forceinline__ float waveReduceAdd(float v) {
  #pragma unroll
  for (int off = 16; off > 0; off >>= 1) v += __shfl_xor(v, off, 32);
  return v;
}

// --- kernel 0: copy T input -> T output region, zero ws accumulators -------
__global__ void k_init(const float* __restrict__ Tin, float* __restrict__ Tout,
                       float* __restrict__ ws, int tsize) {
  int i = blockIdx.x * blockDim.x + threadIdx.x;
  if (i < 64) ws[i] = 0.0f;
  if (i < tsize) Tout[i] = Tin[i];
}

// --- kernel 1: d = <V[j-1], b> ---------------------------------------------
__global__ void k_dot1(const float* __restrict__ b, const float* __restrict__ V,
                       const int* __restrict__ jp, float* __restrict__ ws, int n) {
  int j = *jp;
  const float* v0 = V + (size_t)(j - 1) * (size_t)n;
  size_t tid = (size_t)blockIdx.x * blockDim.x + threadIdx.x;
  size_t stride = (size_t)gridDim.x * blockDim.x;
  float acc = 0.0f;
  for (size_t i = tid * 4; i < (size_t)n; i += stride * 4) {
    float4 bv = *(const float4*)(b + i);
    float4 vv = *(const float4*)(v0 + i);
    acc += bv.x * vv.x + bv.y * vv.y + bv.z * vv.z + bv.w * vv.w;
  }
  acc = waveReduceAdd(acc);
  if ((threadIdx.x & 31) == 0) atomicAdd(&ws[0], acc);
}

// --- kernel 2: b1 = b - d*V[j-1]; d2 = <V[j], b1> --------------------------
__global__ void k_update1_dot2(const float* __restrict__ b, const float* __restrict__ V,
                               const int* __restrict__ jp, float* __restrict__ bout,
                               float* __restrict__ ws, int n) {
  int j = *jp;
  const float* vjm1 = V + (size_t)(j - 1) * (size_t)n;
  const float* vj   = V + (size_t)j * (size_t)n;
  float d = ws[0];
  size_t tid = (size_t)blockIdx.x * blockDim.x + threadIdx.x;
  size_t stride = (size_t)gridDim.x * blockDim.x;
  float acc = 0.0f;
  for (size_t i = tid * 4; i < (size_t)n; i += stride * 4) {
    float4 bv = *(const float4*)(b + i);
    float4 p  = *(const float4*)(vjm1 + i);
    float4 q  = *(const float4*)(vj + i);
    float4 b1;
    b1.x = fmaf(-d, p.x, bv.x); b1.y = fmaf(-d, p.y, bv.y);
    b1.z = fmaf(-d, p.z, bv.z); b1.w = fmaf(-d, p.w, bv.w);
    *(float4*)(bout + i) = b1;
    acc += b1.x * q.x + b1.y * q.y + b1.z * q.z + b1.w * q.w;
  }
  acc = waveReduceAdd(acc);
  if ((threadIdx.x & 31) == 0) atomicAdd(&ws[1], acc);
}

// --- kernel 3: b2 = b1 - d2*V[j] (in-place in bio) and h_i = <V[i], b2>
//     rows 0..15 via V_WMMA_F32_16X16X4_F32 with B = b2 chunk in column 0,
//     built branchlessly: every lane ds_load_b64's its K-pair and scales by a
//     per-lane mask (1.0 only for lanes 0 and 16).  Row j via VALU.
__global__ void k_h_wmma(const float* __restrict__ V, const int* __restrict__ jp,
                         float* __restrict__ bio, float* __restrict__ ws, int n) {
  __shared__ float sb[8][128];
  int j = *jp;
  float d2 = ws[1];
  const float* vj = V + (size_t)j * (size_t)n;
  int lane = threadIdx.x & 31;
  int wave = threadIdx.x >> 5;
  int hi   = (lane >= 16) ? 2 : 0;
  float msk = ((lane & 15) == 0) ? 1.0f : 0.0f;   // lanes 0 and 16 feed column 0
  size_t wgid   = (size_t)blockIdx.x * (blockDim.x >> 5) + wave;
  size_t nwaves = (size_t)gridDim.x * (blockDim.x >> 5);
  size_t nchunks = (size_t)n >> 7;   // 128 elements per chunk

  v8f acc = {0.f, 0.f, 0.f, 0.f, 0.f, 0.f, 0.f, 0.f};
  float hj = 0.0f;
  // A tile base: lanes 0-15 -> (M=lane, K=k0,k0+1); lanes 16-31 ->
  // (M=lane-16, K=k0+2,k0+3)   [32-bit A layout, ISA 7.12.2]
  const float* arowBase = V + (size_t)(lane & 15) * (size_t)n + hi;
  const float* sbw = &sb[wave][hi];

  for (size_t c = wgid; c < nchunks; c += nwaves) {
    size_t base = c * 128;
    size_t idx = base + (size_t)lane * 4;
    float4 b1v = *(const float4*)(bio + idx);
    float4 vjv = *(const float4*)(vj + idx);
    float4 b2v;
    b2v.x = fmaf(-d2, vjv.x, b1v.x); b2v.y = fmaf(-d2, vjv.y, b1v.y);
    b2v.z = fmaf(-d2, vjv.z, b1v.z); b2v.w = fmaf(-d2, vjv.w, b1v.w);
    *(float4*)(bio + idx) = b2v;
    int l4 = lane * 4;
    sb[wave][l4 + 0] = b2v.x; sb[wave][l4 + 1] = b2v.y;
    sb[wave][l4 + 2] = b2v.z; sb[wave][l4 + 3] = b2v.w;
    hj += b2v.x * vjv.x + b2v.y * vjv.y + b2v.z * vjv.z + b2v.w * vjv.w;

    const float* ar = arowBase + base;             // imm-offset loads below
    __builtin_prefetch(ar + nwaves * 128, 0, 1);   // next chunk of this V row
    #pragma unroll
    for (int g = 0; g < 32; ++g) {
      int k0 = g * 4;
      v2f A = *(const v2f*)(ar + k0);              // global_load_b64 off:k0*4
      v2f bl = *(const v2f*)(sbw + k0);            // ds_load_b64 off:(k0+hi)*4
      v2f B; B.x = bl.x * msk; B.y = bl.y * msk;   // zero except lanes 0/16
      acc = __builtin_amdgcn_wmma_f32_16x16x4_f32(
          false, A, false, B, (short)0, acc, false, false);
    }
  }

  hj = waveReduceAdd(hj);
  if (lane == 0) atomicAdd(&ws[24 + j], hj);
  // Column 0 of D: (M=i,N=0) -> i<8: VGPR i lane 0;  i>=8: VGPR i-8 lane 16.
  #pragma unroll
  for (int r = 0; r < 8; ++r) {
    float v0 = __shfl(acc[r], 0, 32);
    float v8 = __shfl(acc[r], 16, 32);
    if (lane == 0) {
      atomicAdd(&ws[24 + r], v0);
      atomicAdd(&ws[24 + 8 + r], v8);
    }
  }
}

// --- kernel 4: w_i = h_i + d*(i==j-1) + d2*(i==j); write T[j-1,j],[j,j-1],[j,j]
__global__ void k_coef(const int* __restrict__ jp, const int* __restrict__ mp,
                       float* __restrict__ ws, float* __restrict__ Tout) {
  if (blockIdx.x == 0 && threadIdx.x == 0) {
    int j = *jp;
    int dim = *mp + 1;
    float d = ws[0], d2 = ws[1];
    for (int i = 0; i < 20; ++i) {
      float wv = (i <= j) ? ws[24 + i] : 0.0f;
      if (i == j - 1) wv += d;
      if (i == j)     wv += d2;
      ws[2 + i] = wv;
    }
    Tout[(size_t)(j - 1) * dim + j] = d;
    Tout[(size_t)j * dim + (j - 1)] = d;
    Tout[(size_t)j * dim + j] = d2;
  }
}

// --- kernel 5: b3 = b2 - sum_{i<20} w_i * V[i];  ss = |b3|^2
//     5 chained V_WMMA_F32_16X16X4_F32 per 16-element chunk:
//     A = broadcast(w k-group), B = 4 rows x 16 cols of V (coalesced b32).
__global__ void k_update_wmma(const float* __restrict__ V, float* __restrict__ bio,
                              float* __restrict__ ws, int n) {
  int lane = threadIdx.x & 31;
  int wave = threadIdx.x >> 5;
  int col = lane & 15;
  int hi  = (lane >= 16) ? 2 : 0;
  float w[20];
  #pragma unroll
  for (int i = 0; i < 20; ++i) w[i] = ws[2 + i];

  // Per-lane row pointers for the 10 V rows this lane streams.
  const float* rowp[10];
  #pragma unroll
  for (int g = 0; g < 5; ++g) {
    rowp[2 * g]     = V + (size_t)(g * 4 + hi) * (size_t)n;
    rowp[2 * g + 1] = V + (size_t)(g * 4 + hi + 1) * (size_t)n;
  }

  size_t wgid   = (size_t)blockIdx.x * (blockDim.x >> 5) + wave;
  size_t nwaves = (size_t)gridDim.x * (blockDim.x >> 5);
  size_t nchunks = (size_t)n >> 4;   // 16 elements per chunk
  float ssp = 0.0f;

  for (size_t c = wgid; c < nchunks; c += nwaves) {
    size_t o = c * 16 + (size_t)col;
    float bv = bio[c * 16 + (size_t)col];          // all lanes load (dups hit L0)
    __builtin_prefetch(rowp[0] + o + nwaves * 16, 0, 1);
    v8f acc = {0.f, 0.f, 0.f, 0.f, 0.f, 0.f, 0.f, 0.f};
    #pragma unroll
    for (int g = 0; g < 5; ++g) {
      v2f A;                            // A[M][K] = w[g*4+K], all rows identical
      A.x = w[g * 4 + hi]; A.y = w[g * 4 + hi + 1];
      v2f B;                            // B[K][N] = V[g*4+K][base+N]
      B.x = rowp[2 * g][o];
      B.y = rowp[2 * g + 1][o];
      acc = __builtin_amdgcn_wmma_f32_16x16x4_f32(
          false, A, false, B, (short)0, acc, false, false);
    }
    float bo = bv - acc[0];             // D row M=0: VGPR0, N=lane (dup in 16-31)
    if (lane < 16) {
      bio[c * 16 + (size_t)col] = bo;
      ssp += bo * bo;
    }
  }
  ssp = waveReduceAdd(ssp);
  if (lane == 0) atomicAdd(&ws[22], ssp);
}

// --- kernel 6: beta, scale, T[j,j+1], T[j+1,j] ------------------------------
__global__ void k_fin(const int* __restrict__ jp, const int* __restrict__ mp,
                      float* __restrict__ ws, float* __restrict__ Tout,
                      float* __restrict__ beta_out) {
  if (blockIdx.x == 0 && threadIdx.x == 0) {
    int j = *jp;
    int dim = *mp + 1;
    float beta = sqrtf(ws[22]);
    bool pos = beta > 0.0f;
    ws[23] = pos ? (1.0f / beta) : 1.0f;
    *beta_out = beta;
    if (pos) {
      Tout[(size_t)j * dim + (j + 1)] = beta;
      Tout[(size_t)(j + 1) * dim + j] = beta;
    }
  }
}

// --- kernel 7: b *= scale ---------------------------------------------------
__global__ void k_scale(float* __restrict__ bio, const float* __restrict__ ws, int n) {
  float s = ws[23];
  size_t tid = (size_t)blockIdx.x * blockDim.x + threadIdx.x;
  size_t stride = (size_t)gridDim.x * blockDim.x;
  for (size_t i = tid * 4; i < (size_t)n; i += stride * 4) {
    float4 v = *(const float4*)(bio + i);
    v.x *= s; v.y *= s; v.z *= s; v.w *= s;
    *(float4*)(bio + i) = v;
  }
}

extern "C" void kernel_launch(void* const* d_in, const int* in_sizes, int n_in,
                              void* d_out, int out_size, void* d_ws, size_t ws_size,
                              hipStream_t stream) {
  const float* b   = (const float*)d_in[0];
  const float* V   = (const float*)d_in[1];
  const float* Tin = (const float*)d_in[2];
  const int*   jp  = (const int*)d_in[3];
  const int*   mp  = (const int*)d_in[4];
  int n     = in_sizes[0];        // 4194304
  int tsize = in_sizes[2];        // (M+1)^2 = 1089

  float* bout     = (float*)d_out;
  float* Tout     = bout + n;
  float* beta_out = Tout + tsize;
  float* ws       = (float*)d_ws;

  k_init<<<(tsize + 255) / 256, 256, 0, stream>>>(Tin, Tout, ws, tsize);
  k_dot1<<<1024, 256, 0, stream>>>(b, V, jp, ws, n);
  k_update1_dot2<<<1024, 256, 0, stream>>>(b, V, jp, bout, ws, n);
  k_h_wmma<<<1024, 256, 0, stream>>>(V, jp, bout, ws, n);
  k_coef<<<1, 64, 0, stream>>>(jp, mp, ws, Tout);
  k_update_wmma<<<1024, 256, 0, stream>>>(V, bout, ws, n);
  k_fin<<<1, 64, 0, stream>>>(jp, mp, ws, Tout, beta_out);
  k_scale<<<1024, 256, 0, stream>>>(bout, ws, n);
}